// GraphColoringPolicy_20212116095331
// MI455X (gfx1250) — compile-verified
//
#include <hip/hip_runtime.h>

// GCN forward for MI455X (gfx1250, wave32).
// GEMMs use V_WMMA_F32_16X16X4_F32 (fp32 precision preserved, matrix pipe used).
// Edge gather/scatter is L2-resident (feature tables = 25.6MB << 192MB L2),
// done with coalesced float2 gathers + native global_atomic_add_f32 (no CAS loop).

typedef float v2f __attribute__((ext_vector_type(2)));
typedef float v8f __attribute__((ext_vector_type(8)));

#define NEG_INF_F (-3.4028234663852886e38f)   // jnp.finfo(float32).min == -FLT_MAX

// Native hardware fp32 atomic add, agent scope, relaxed (no return needed ->
// atomic-without-return, tracked on STOREcnt; avoids CAS-loop lowering).
__device__ __forceinline__ void atomAddF(float* p, float v) {
  __hip_atomic_fetch_add(p, v, __ATOMIC_RELAXED, __HIP_MEMORY_SCOPE_AGENT);
}

// ---------------- degree / dinv ----------------
__global__ void k_deg_init(float* __restrict__ deg, int n) {
  int i = blockIdx.x * blockDim.x + threadIdx.x;
  if (i < n) deg[i] = 1.0f;                    // self-loop contributes 1
}
__global__ void k_deg_count(const int* __restrict__ dst, float* __restrict__ deg, int nE) {
  int e = blockIdx.x * blockDim.x + threadIdx.x;
  if (e < nE) atomAddF(&deg[dst[e]], 1.0f);
}
__global__ void k_deg_rsqrt(float* __restrict__ deg, int n) {
  int i = blockIdx.x * blockDim.x + threadIdx.x;
  if (i < n) deg[i] = rsqrtf(deg[i]);
}

// ---------------- zero workspace ----------------
__global__ void k_zero(float4* __restrict__ p, int n4) {
  int i = blockIdx.x * blockDim.x + threadIdx.x;
  if (i < n4) p[i] = make_float4(0.f, 0.f, 0.f, 0.f);
}

// ---------------- C[M x Ncols] = A[M x K] * B[K x Ncols], fp32 WMMA ----------------
// One wave per 16x16 tile. blockIdx.x = row tile, threadIdx.y = col tile.
// A-frag (16x4 f32): lanes 0-15 hold K={k0,k0+1}, lanes 16-31 hold K={k0+2,k0+3}.
// C/D (16x16 f32): vgpr r, lanes 0-15 -> M=r, lanes 16-31 -> M=r+8.
__global__ void k_gemm_wmma(const float* __restrict__ A, const float* __restrict__ B,
                            float* __restrict__ C, int K, int Ncols) {
  const int lane    = threadIdx.x & 31;
  const int half    = lane >> 4;               // 0: lanes 0-15, 1: lanes 16-31
  const int l15     = lane & 15;
  const int rowBase = blockIdx.x * 16;
  const int colBase = threadIdx.y * 16;

  const float* arow = A + (size_t)(rowBase + l15) * K;
  const float* bcol = B + colBase + l15;

  v8f acc = {0.f, 0.f, 0.f, 0.f, 0.f, 0.f, 0.f, 0.f};
  for (int k0 = 0; k0 < K; k0 += 4) {
    const int ka = k0 + half * 2;
    v2f a, b;
    a.x = arow[ka];
    a.y = arow[ka + 1];
    b.x = bcol[(size_t)ka * Ncols];
    b.y = bcol[(size_t)(ka + 1) * Ncols];
    acc = __builtin_amdgcn_wmma_f32_16x16x4_f32(false, a, false, b, (short)0, acc,
                                                false, false);
  }
  float* crow = C + (size_t)(rowBase + half * 8) * Ncols + colBase + l15;
#pragma unroll
  for (int r = 0; r < 8; ++r) crow[(size_t)r * Ncols] = acc[r];
}

// ---------------- edge scatter: agg[dst] += hW[src] * dinv[src]*dinv[dst] ----------------
// One wave per edge; H=64 columns -> each lane handles a contiguous float2.
__global__ void k_scatter(const int* __restrict__ src, const int* __restrict__ dst,
                          const float* __restrict__ dinv, const float* __restrict__ hW,
                          float* __restrict__ agg, int nE) {
  const int lane = threadIdx.x & 31;
  const int wave = (blockIdx.x * blockDim.x + threadIdx.x) >> 5;
  if (wave >= nE) return;
  const int s = src[wave];
  const int d = dst[wave];
  const float w = dinv[s] * dinv[d];
  const float2 v = ((const float2*)(hW + (size_t)s * 64))[lane];
  float* ap = agg + (size_t)d * 64 + lane * 2;
  atomAddF(ap,     v.x * w);
  atomAddF(ap + 1, v.y * w);
}

// ---------------- out = relu(agg + hW*dinv^2 (self-loop) + bias) ----------------
__global__ void k_bias_relu_selfloop(const float* __restrict__ agg, const float* __restrict__ hW,
                                     const float* __restrict__ dinv, const float* __restrict__ bias,
                                     float* __restrict__ out, int n) {
  int idx = blockIdx.x * blockDim.x + threadIdx.x;
  if (idx >= n) return;
  const int i = idx >> 6;          // H = 64
  const int j = idx & 63;
  const float di = dinv[i];
  const float v = agg[idx] + hW[idx] * di * di + bias[j];
  out[idx] = fmaxf(v, 0.0f);
}

// ---------------- head: logits = H @ Wh + bh, masked. N=10 padded to 16. ----------------
__global__ void k_head_wmma(const float* __restrict__ H, const float* __restrict__ Wh,
                            const float* __restrict__ bh, const int* __restrict__ mask,
                            float* __restrict__ out, int K, int C) {
  const int lane    = threadIdx.x & 31;
  const int half    = lane >> 4;
  const int l15     = lane & 15;
  const int rowBase = blockIdx.x * 16;
  const bool valid  = (l15 < C);

  const float* arow = H + (size_t)(rowBase + l15) * K;
  v8f acc = {0.f, 0.f, 0.f, 0.f, 0.f, 0.f, 0.f, 0.f};
  for (int k0 = 0; k0 < K; k0 += 4) {
    const int ka = k0 + half * 2;
    v2f a, b;
    a.x = arow[ka];
    a.y = arow[ka + 1];
    b.x = valid ? Wh[ka * C + l15]       : 0.0f;   // zero-pad columns C..15
    b.y = valid ? Wh[(ka + 1) * C + l15] : 0.0f;
    acc = __builtin_amdgcn_wmma_f32_16x16x4_f32(false, a, false, b, (short)0, acc,
                                                false, false);
  }
  if (valid) {
    const float bias = bh[l15];
    const int row0 = rowBase + half * 8;
#pragma unroll
    for (int r = 0; r < 8; ++r) {
      const int idx = (row0 + r) * C + l15;
      out[idx] = mask[idx] ? (acc[r] + bias) : NEG_INF_F;
    }
  }
}

extern "C" void kernel_launch(void* const* d_in, const int* in_sizes, int n_in,
                              void* d_out, int out_size, void* d_ws, size_t ws_size,
                              hipStream_t stream) {
  (void)n_in; (void)out_size; (void)ws_size;
  const float* x   = (const float*)d_in[0];
  const int*   ei  = (const int*)  d_in[1];   // [2, E] flat: row 0 = src, row 1 = dst
  const int*   msk = (const int*)  d_in[2];
  const float* W1  = (const float*)d_in[3];
  const float* b1  = (const float*)d_in[4];
  const float* W2  = (const float*)d_in[5];
  const float* b2  = (const float*)d_in[6];
  const float* Wh  = (const float*)d_in[7];
  const float* bh  = (const float*)d_in[8];
  float*       out = (float*)d_out;

  const int Hd = in_sizes[4];            // 64
  const int C  = in_sizes[8];            // 10
  const int IN = in_sizes[3] / Hd;       // 128
  const int N  = in_sizes[0] / IN;       // 100000 (divisible by 16)
  const int E  = in_sizes[1] / 2;        // 3200000

  const int* src = ei;
  const int* dst = ei + E;

  // workspace: dinv[N] | buf0[N*H] | buf1[N*H]   (~52 MB)
  float* dinv = (float*)d_ws;
  float* buf0 = dinv + (((size_t)N + 255) & ~(size_t)255);
  float* buf1 = buf0 + (size_t)N * Hd;

  const int T = 256;
  const int nH = N * Hd;
  const int rowTiles = N / 16;
  const dim3 gemmBlk(32, Hd / 16);       // 4 waves/block, one 16x16 tile each

  // degree -> dinv
  k_deg_init <<<(N + T - 1) / T, T, 0, stream>>>(dinv, N);
  k_deg_count<<<(E + T - 1) / T, T, 0, stream>>>(dst, dinv, E);
  k_deg_rsqrt<<<(N + T - 1) / T, T, 0, stream>>>(dinv, N);

  // ---- layer 1 ----
  k_gemm_wmma<<<rowTiles, gemmBlk, 0, stream>>>(x, W1, buf0, IN, Hd);
  k_zero<<<(nH / 4 + T - 1) / T, T, 0, stream>>>((float4*)buf1, nH / 4);
  k_scatter<<<(E + 7) / 8, T, 0, stream>>>(src, dst, dinv, buf0, buf1, E);
  k_bias_relu_selfloop<<<(nH + T - 1) / T, T, 0, stream>>>(buf1, buf0, dinv, b1, buf0, nH);

  // ---- layer 2 ----
  k_gemm_wmma<<<rowTiles, gemmBlk, 0, stream>>>(buf0, W2, buf1, Hd, Hd);
  k_zero<<<(nH / 4 + T - 1) / T, T, 0, stream>>>((float4*)buf0, nH / 4);
  k_scatter<<<(E + 7) / 8, T, 0, stream>>>(src, dst, dinv, buf1, buf0, E);
  k_bias_relu_selfloop<<<(nH + T - 1) / T, T, 0, stream>>>(buf0, buf1, dinv, b2, buf0, nH);

  // ---- head + mask ----
  k_head_wmma<<<rowTiles, 32, 0, stream>>>(buf0, Wh, bh, msk, out, Hd, C);
}